// BronxModel_35072702939161
// MI455X (gfx1250) — compile-verified
//
#include <hip/hip_runtime.h>
#include <hip/hip_bf16.h>

#define N_NODES 100000
#define N_EDGES 1600000
#define IN_F    128
#define HID_F   64
#define OUT_F   40
#define N_STEPS 10
#define DT      0.1f
#define SIG_SQDT 0.0031622776601683794f   // SIGMA * sqrt(DT) = 0.01 * sqrt(0.1)

typedef __attribute__((ext_vector_type(2))) float v2f;
typedef __attribute__((ext_vector_type(8))) float v8f;

// ---------------------------------------------------------------------------
// degree / norm precompute
// ---------------------------------------------------------------------------
__global__ void zero_kernel(float* __restrict__ p, int n) {
    int i = blockIdx.x * blockDim.x + threadIdx.x;
    if (i < n) p[i] = 0.0f;
}

__global__ void degree_kernel(const int* __restrict__ src, const int* __restrict__ dst,
                              float* __restrict__ deg_out, float* __restrict__ deg_in, int e) {
    int i = blockIdx.x * blockDim.x + threadIdx.x;
    if (i < e) {
        atomicAdd(&deg_out[src[i]], 1.0f);
        atomicAdd(&deg_in[dst[i]], 1.0f);
    }
}

__global__ void norm_kernel(const int* __restrict__ src, const int* __restrict__ dst,
                            const float* __restrict__ deg_out, const float* __restrict__ deg_in,
                            float* __restrict__ norm, int e) {
    int i = blockIdx.x * blockDim.x + threadIdx.x;
    if (i < e) {
        float doo = fmaxf(deg_out[src[i]], 1.0f);
        float din = fmaxf(deg_in[dst[i]], 1.0f);
        norm[i] = rsqrtf(doo) * rsqrtf(din);
    }
}

// Zero-pad W_out [64 x 40] into Bp [64 x 48] so the GEMM inner loop needs no
// column guard; only the store epilogue guards col < 40.
__global__ void pad_wout_kernel(const float* __restrict__ W_out, float* __restrict__ Bp) {
    int i = blockIdx.x * blockDim.x + threadIdx.x;   // over 64*48
    if (i < HID_F * 48) {
        int r = i / 48, c = i % 48;
        Bp[i] = (c < OUT_F) ? W_out[r * OUT_F + c] : 0.0f;
    }
}

// ---------------------------------------------------------------------------
// FP32 WMMA GEMM: C[M x LDC] = A[M x K] * B[K x N], row-major, exact f32.
// One wave32 per 16 x N row-tile (N/16 accumulators, A fragment reused across
// all column tiles); grid = M/16, blockDim = 32. K, N compile-time so all B
// addresses fold to base + immediate — no guards, no 64-bit math in the loop.
// A 16x4 fragment (ISA 7.12.2): lanes 0-15 carry K={k0,k0+1} of rows 0..15,
// lanes 16-31 carry K={k0+2,k0+3}. B 4x16: K-row striped across lanes with
// the same half-wave pairing. C/D: VGPR i = row i (lanes 0-15) / i+8 (16-31).
// ---------------------------------------------------------------------------
template <int K, int N, int LDC>
__global__ void wmma_gemm_kernel(const float* __restrict__ A,
                                 const float* __restrict__ B,
                                 float* __restrict__ C) {
    constexpr int NT = N / 16;
    const int lane    = threadIdx.x & 31;
    const int half    = lane >> 4;     // 0: lanes 0-15, 1: lanes 16-31
    const int lrow    = lane & 15;
    const int rowBase = blockIdx.x * 16;

    const float* __restrict__ arow  = A + (size_t)(rowBase + lrow) * K + 2 * half;
    const float* __restrict__ bbase = B + (size_t)(2 * half) * N + lrow;

    v8f acc[NT] = {};

#pragma unroll
    for (int k0 = 0; k0 < K; k0 += 4) {
        v2f a;
        a.x = arow[k0];
        a.y = arow[k0 + 1];
#pragma unroll
        for (int t = 0; t < NT; ++t) {
            v2f b;
            b.x = bbase[k0 * N + t * 16];
            b.y = bbase[(k0 + 1) * N + t * 16];
            acc[t] = __builtin_amdgcn_wmma_f32_16x16x4_f32(
                /*neg_a=*/false, a, /*neg_b=*/false, b,
                /*c_mod=*/(short)0, acc[t], /*reuse_a=*/false, /*reuse_b=*/false);
        }
    }

#pragma unroll
    for (int t = 0; t < NT; ++t) {
        const int col = t * 16 + lrow;
        if (N == LDC || col < LDC) {
#pragma unroll
            for (int i = 0; i < 8; ++i) {
                C[(size_t)(rowBase + i + 8 * half) * LDC + col] = acc[t][i];
            }
        }
    }
}

// ---------------------------------------------------------------------------
// x_next = (1-DT)*x + SIGMA*sqrt(dt)*dW_k   (fused Euler-Maruyama local part,
// also serves as the zero-free scatter accumulator initialization)
// ---------------------------------------------------------------------------
__global__ void prestep_kernel(const float4* __restrict__ x,
                               const float4* __restrict__ dwk,
                               float4* __restrict__ x_next, int n4) {
    int i = blockIdx.x * blockDim.x + threadIdx.x;
    if (i < n4) {
        float4 xv = x[i];
        float4 wv = dwk[i];
        float4 r;
        r.x = (1.0f - DT) * xv.x + SIG_SQDT * wv.x;
        r.y = (1.0f - DT) * xv.y + SIG_SQDT * wv.y;
        r.z = (1.0f - DT) * xv.z + SIG_SQDT * wv.z;
        r.w = (1.0f - DT) * xv.w + SIG_SQDT * wv.w;
        x_next[i] = r;
    }
}

// ---------------------------------------------------------------------------
// Edge scatter: one wave32 per edge; each lane handles a float2 (64 feats).
// x_next[dst] += DT * norm[e] * m[src] via global f32 atomics (node arrays
// are 25.6 MB << 192 MB L2, so gather/scatter traffic stays L2-resident).
// ---------------------------------------------------------------------------
__global__ void scatter_kernel(const float* __restrict__ m,
                               const float* __restrict__ norm,
                               const int* __restrict__ src,
                               const int* __restrict__ dst,
                               float* __restrict__ x_next, int n_edges) {
    long long gid = (long long)blockIdx.x * blockDim.x + threadIdx.x;
    int e = (int)(gid >> 5);
    if (e >= n_edges) return;
    int lane = (int)(gid & 31);
    int s = src[e];
    int d = dst[e];
    float w = DT * norm[e];
    const float2* ms = (const float2*)(m + (size_t)s * HID_F);
    float2 v = ms[lane];
    float* xd = x_next + (size_t)d * HID_F + lane * 2;
    atomicAdd(xd,     w * v.x);
    atomicAdd(xd + 1, w * v.y);
}

// ---------------------------------------------------------------------------
extern "C" void kernel_launch(void* const* d_in, const int* in_sizes, int n_in,
                              void* d_out, int out_size, void* d_ws, size_t ws_size,
                              hipStream_t stream) {
    const float* h     = (const float*)d_in[0];
    const float* W_in  = (const float*)d_in[1];
    const float* W_msg = (const float*)d_in[2];
    const float* W_out = (const float*)d_in[3];
    const float* dW    = (const float*)d_in[4];
    const int*   src   = (const int*)d_in[5];
    const int*   dst   = (const int*)d_in[6];
    float*       out   = (float*)d_out;

    // workspace layout (floats)
    const size_t NH = (size_t)N_NODES * HID_F;   // 6.4M floats
    float* P0      = (float*)d_ws;               // x ping
    float* M1      = P0 + NH;                    // message buffer
    float* P2      = M1 + NH;                    // x pong
    float* deg_out = P2 + NH;
    float* deg_in  = deg_out + N_NODES;
    float* norm    = deg_in + N_NODES;           // [E]
    float* Bp      = norm + N_EDGES;             // padded W_out [64 x 48]

    const int threads = 256;

    // --- degrees + edge norms + padded W_out ---
    zero_kernel<<<(2 * N_NODES + threads - 1) / threads, threads, 0, stream>>>(deg_out, 2 * N_NODES);
    degree_kernel<<<(N_EDGES + threads - 1) / threads, threads, 0, stream>>>(src, dst, deg_out, deg_in, N_EDGES);
    norm_kernel<<<(N_EDGES + threads - 1) / threads, threads, 0, stream>>>(src, dst, deg_out, deg_in, norm, N_EDGES);
    pad_wout_kernel<<<(HID_F * 48 + threads - 1) / threads, threads, 0, stream>>>(W_out, Bp);

    // --- x = h @ W_in ---
    wmma_gemm_kernel<IN_F, HID_F, HID_F><<<N_NODES / 16, 32, 0, stream>>>(h, W_in, P0);

    // --- 10 Euler-Maruyama steps ---
    float* cur = P0;
    float* nxt = P2;
    const int n4 = (int)(NH / 4);
    for (int k = 0; k < N_STEPS; ++k) {
        wmma_gemm_kernel<HID_F, HID_F, HID_F><<<N_NODES / 16, 32, 0, stream>>>(cur, W_msg, M1);
        prestep_kernel<<<(n4 + threads - 1) / threads, threads, 0, stream>>>(
            (const float4*)cur, (const float4*)(dW + (size_t)k * NH), (float4*)nxt, n4);
        long long swork = (long long)N_EDGES * 32;
        scatter_kernel<<<(unsigned)((swork + threads - 1) / threads), threads, 0, stream>>>(
            M1, norm, src, dst, nxt, N_EDGES);
        float* t = cur; cur = nxt; nxt = t;
    }

    // --- out = x @ W_out: padded 64x48 B, store guard only at epilogue ---
    wmma_gemm_kernel<HID_F, 48, OUT_F><<<N_NODES / 16, 32, 0, stream>>>(cur, Bp, out);
}